// Decoder3_68848325755388
// MI455X (gfx1250) — compile-verified
//
#include <hip/hip_runtime.h>
#include <hip/hip_bf16.h>
#include <math.h>

// Problem constants (match reference)
#define BB 128
#define SS 512
#define HH 768
#define NLAB 42
#define LDIM 300
#define DKK 64
#define FEAT (3 * HH)     // 2304
#define PP 512
#define NEGV (-1.0e9f)
#define NINF (-1.0e12f)   // reference INF = 1e12

typedef float v2f __attribute__((ext_vector_type(2)));
typedef float v8f __attribute__((ext_vector_type(8)));

__device__ __forceinline__ float gelu_f(float x) {
    return 0.5f * x * (1.0f + erff(x * 0.70710678118654752f));
}

// ---------------------------------------------------------------------------
// Kernel 0: kvec = einsum('ld,lkd->lk', lab, WK)+bK ; w_eff = einsum('lk,lkh->h');
//           c = sum(kvec*bQ).  All tiny — one block.
// ---------------------------------------------------------------------------
__global__ __launch_bounds__(256) void prep_kernel(
    const float* __restrict__ label_emb,  // (42,300)
    const float* __restrict__ WK,         // (41,64,300)
    const float* __restrict__ bK,         // (41,64)
    const float* __restrict__ WQ,         // (41,64,768)
    const float* __restrict__ bQ,         // (41,64)
    float* __restrict__ w_eff,            // (768)
    float* __restrict__ c_out)            // (1)
{
    __shared__ float kv[(NLAB - 1) * DKK];   // 2624 floats
    __shared__ float red[256];
    const int tid = threadIdx.x;
    const int NK = (NLAB - 1) * DKK;

    for (int i = tid; i < NK; i += 256) {
        int l = i >> 6;
        const float* lv = label_emb + (size_t)(1 + l) * LDIM;
        const float* wk = WK + (size_t)i * LDIM;
        float s = 0.0f;
        for (int d = 0; d < LDIM; ++d) s += lv[d] * wk[d];
        kv[i] = s + bK[i];
    }
    __syncthreads();

    for (int h = tid; h < HH; h += 256) {
        float s = 0.0f;
        for (int i = 0; i < NK; ++i) s += kv[i] * WQ[(size_t)i * HH + h];
        w_eff[h] = s;
    }

    float s = 0.0f;
    for (int i = tid; i < NK; i += 256) s += kv[i] * bQ[i];
    red[tid] = s;
    __syncthreads();
    for (int off = 128; off; off >>= 1) {
        if (tid < off) red[tid] += red[tid + off];
        __syncthreads();
    }
    if (tid == 0) *c_out = red[0];
}

// ---------------------------------------------------------------------------
// Kernel 1: per-batch fused scores -> softmax -> attention pooling + masked
// maxes.  One block per batch, 256 threads (8 wave32s).
// pooled[b,h] = sum_s attn[b,s]*inputs[b,s,h]
// feats[b, 0:768]    = max_s (subj_pos==P ? x : -1e12)
// feats[b, 768:1536] = max_s (obj_pos ==P ? x : -1e12)
// ---------------------------------------------------------------------------
__global__ __launch_bounds__(256) void attn_pool_kernel(
    const float* __restrict__ inputs,
    const int* __restrict__ pos_ids,
    const int* __restrict__ subj_pos,
    const int* __restrict__ obj_pos,
    const float* __restrict__ w_eff,
    const float* __restrict__ c_ptr,
    float* __restrict__ pooled,      // (128,768)
    float* __restrict__ feats)       // (128,2304)
{
    const int b = blockIdx.x;
    const int tid = threadIdx.x;
    const int lane = tid & 31;
    const int wave = tid >> 5;

    __shared__ float sw[HH];
    __shared__ float sc[SS];
    __shared__ float fsub[SS];
    __shared__ float fobj[SS];
    __shared__ float red[8];

    for (int h = tid; h < HH; h += 256) sw[h] = w_eff[h];
    const float cval = *c_ptr;

    for (int s = tid; s < SS; s += 256) {
        const int idx = b * SS + s;
        const int sp = subj_pos[idx];
        const int op = obj_pos[idx];
        const int pid = pos_ids[idx];
        const bool m = (sp == PP) || (op == PP) || (pid == 0);
        sc[s] = m ? (float)(NLAB - 1) * NEGV : 0.0f;   // mask additive term
        fsub[s] = (sp == PP) ? 1.0f : 0.0f;
        fobj[s] = (op == PP) ? 1.0f : 0.0f;
    }
    __syncthreads();

    // wave-cooperative dot products: wave w owns rows s = w, w+8, ...
    for (int s = wave; s < SS; s += 8) {
        const float* xr = inputs + ((size_t)b * SS + s) * HH;
        float d = 0.0f;
        for (int h = lane; h < HH; h += 32) d += xr[h] * sw[h];
        for (int off = 16; off; off >>= 1) d += __shfl_xor(d, off, 32);
        if (lane == 0) sc[s] = (d + cval) * 0.125f + sc[s];
    }
    __syncthreads();

    // softmax over S=512 (each thread owns 2 entries)
    float mx = fmaxf(sc[tid], sc[tid + 256]);
    for (int off = 16; off; off >>= 1) mx = fmaxf(mx, __shfl_xor(mx, off, 32));
    if (lane == 0) red[wave] = mx;
    __syncthreads();
    float bmax = red[0];
    for (int w = 1; w < 8; ++w) bmax = fmaxf(bmax, red[w]);
    __syncthreads();

    float e0 = __expf(sc[tid] - bmax);
    float e1 = __expf(sc[tid + 256] - bmax);
    sc[tid] = e0;
    sc[tid + 256] = e1;
    float sum = e0 + e1;
    for (int off = 16; off; off >>= 1) sum += __shfl_xor(sum, off, 32);
    if (lane == 0) red[wave] = sum;
    __syncthreads();
    float bsum = 0.0f;
    for (int w = 0; w < 8; ++w) bsum += red[w];
    const float inv = 1.0f / bsum;
    __syncthreads();

    // fused pooling + masked maxes: thread owns columns tid, tid+256, tid+512
    const int h0 = tid, h1 = tid + 256, h2 = tid + 512;
    float p0 = 0.f, p1 = 0.f, p2 = 0.f;
    float sm0 = NINF, sm1 = NINF, sm2 = NINF;
    float om0 = NINF, om1 = NINF, om2 = NINF;
    for (int s = 0; s < SS; ++s) {
        const float* xr = inputs + ((size_t)b * SS + s) * HH;
        const float a = sc[s] * inv;
        const float fs = fsub[s];
        const float fo = fobj[s];
        const float x0 = xr[h0], x1 = xr[h1], x2 = xr[h2];
        p0 += a * x0; p1 += a * x1; p2 += a * x2;
        if (fs != 0.0f) { sm0 = fmaxf(sm0, x0); sm1 = fmaxf(sm1, x1); sm2 = fmaxf(sm2, x2); }
        if (fo != 0.0f) { om0 = fmaxf(om0, x0); om1 = fmaxf(om1, x1); om2 = fmaxf(om2, x2); }
    }
    float* pr = pooled + (size_t)b * HH;
    pr[h0] = p0; pr[h1] = p1; pr[h2] = p2;
    float* fr = feats + (size_t)b * FEAT;
    fr[h0] = sm0; fr[h1] = sm1; fr[h2] = sm2;
    fr[HH + h0] = om0; fr[HH + h1] = om1; fr[HH + h2] = om2;
}

// ---------------------------------------------------------------------------
// WMMA fp32 GEMM: D = act(A(MxK) @ W(NxK)^T + bias), row-major everywhere.
// One 16x16 D tile per wave via V_WMMA_F32_16X16X4_F32; block = 4 waves = 64
// output columns.
//
// N-edge handling (N=300): output column n depends only on B-fragment data in
// the lanes with lane&15 == n-n0 — the exact lanes whose stores we skip.  So
// out-of-range lanes may compute garbage; we only clamp their load address to
// a valid row and guard the store.  No data zeroing in the inner loop.
//
// A-frag 16x4 f32 (ISA 7.12.2): v0 = K {0|2} by lane half, v1 = K {1|3}.
// B-frag 4x16: v0 = row K {0|2}, v1 = row K {1|3}, col = lane&15.
// C/D: vgpr r -> row r (lanes 0-15) / r+8 (lanes 16-31), col = lane&15.
// ---------------------------------------------------------------------------
template <int ACT>
__global__ __launch_bounds__(128) void gemm_wmma_f32(
    const float* __restrict__ A, const float* __restrict__ W,
    const float* __restrict__ bias, float* __restrict__ D,
    int M, int N, int K)
{
    const int wave = threadIdx.x >> 5;
    const int lane = threadIdx.x & 31;
    const int half = lane >> 4;
    const int lr = lane & 15;
    const int n0 = blockIdx.x * 64 + wave * 16;
    const int m0 = blockIdx.y * 16;
    if (n0 >= N) return;                 // wave-uniform -> EXEC stays all-1s

    const int ncol = n0 + lr;
    const bool nvalid = (ncol < N);

    const float2* a2 =
        reinterpret_cast<const float2*>(A + (size_t)(m0 + lr) * K + 2 * half);
    const float2* b2 =
        reinterpret_cast<const float2*>(W + (size_t)(nvalid ? ncol : n0) * K + 2 * half);

    v8f acc = {};
    for (int k0 = 0; k0 < K; k0 += 4) {
        const float2 av = a2[k0 >> 1];
        const float2 bv = b2[k0 >> 1];
        v2f a;  a[0] = av.x;  a[1] = av.y;
        v2f bb; bb[0] = bv.x; bb[1] = bv.y;
        acc = __builtin_amdgcn_wmma_f32_16x16x4_f32(
            /*neg_a=*/false, a, /*neg_b=*/false, bb,
            /*c_mod=*/(short)0, acc, /*reuse_a=*/false, /*reuse_b=*/false);
    }

    if (nvalid) {
        const float bs = bias[ncol];
        #pragma unroll
        for (int r = 0; r < 8; ++r) {
            const int row = m0 + r + 8 * half;
            float v = acc[r] + bs;
            if (ACT) v = gelu_f(v);
            D[(size_t)row * N + ncol] = v;
        }
    }
}

// ---------------------------------------------------------------------------
// LayerNorm over ctx rows; writes d_out ctx section AND feats[:,1536:2304].
// ---------------------------------------------------------------------------
__global__ __launch_bounds__(256) void ln_kernel(
    const float* __restrict__ X,       // (128,768) gelu'd ctx
    const float* __restrict__ g, const float* __restrict__ bt,
    float* __restrict__ ctx_out,       // d_out + 128*42
    float* __restrict__ feats)
{
    const int b = blockIdx.x;
    const int tid = threadIdx.x;
    const int lane = tid & 31;
    const int wave = tid >> 5;
    __shared__ float red[8];

    const float* xr = X + (size_t)b * HH;
    const float x0 = xr[tid], x1 = xr[tid + 256], x2 = xr[tid + 512];

    float s = x0 + x1 + x2;
    for (int off = 16; off; off >>= 1) s += __shfl_xor(s, off, 32);
    if (lane == 0) red[wave] = s;
    __syncthreads();
    float tot = 0.0f;
    for (int w = 0; w < 8; ++w) tot += red[w];
    const float mu = tot * (1.0f / HH);
    __syncthreads();

    const float d0 = x0 - mu, d1 = x1 - mu, d2 = x2 - mu;
    float sq = d0 * d0 + d1 * d1 + d2 * d2;
    for (int off = 16; off; off >>= 1) sq += __shfl_xor(sq, off, 32);
    if (lane == 0) red[wave] = sq;
    __syncthreads();
    float vtot = 0.0f;
    for (int w = 0; w < 8; ++w) vtot += red[w];
    const float inv = rsqrtf(vtot * (1.0f / HH) + 1e-5f);

    float* co = ctx_out + (size_t)b * HH;
    float* fr = feats + (size_t)b * FEAT + 2 * HH;
    const float y0 = d0 * inv * g[tid] + bt[tid];
    const float y1 = d1 * inv * g[tid + 256] + bt[tid + 256];
    const float y2 = d2 * inv * g[tid + 512] + bt[tid + 512];
    co[tid] = y0; co[tid + 256] = y1; co[tid + 512] = y2;
    fr[tid] = y0; fr[tid + 256] = y1; fr[tid + 512] = y2;
}

// ---------------------------------------------------------------------------
// logits = h2(128,300) @ label_emb(42,300)^T -> d_out[0:128*42]   (tiny)
// ---------------------------------------------------------------------------
__global__ __launch_bounds__(64) void logits_kernel(
    const float* __restrict__ h2, const float* __restrict__ lemb,
    float* __restrict__ out)
{
    const int b = blockIdx.x;
    const int l = threadIdx.x;
    if (l >= NLAB) return;
    const float* f = h2 + (size_t)b * LDIM;
    const float* e = lemb + (size_t)l * LDIM;
    float s = 0.0f;
    for (int d = 0; d < LDIM; ++d) s += f[d] * e[d];
    out[b * NLAB + l] = s;
}

// ---------------------------------------------------------------------------
extern "C" void kernel_launch(void* const* d_in, const int* in_sizes, int n_in,
                              void* d_out, int out_size, void* d_ws, size_t ws_size,
                              hipStream_t stream) {
    const float* inputs    = (const float*)d_in[0];
    const int*   pos_ids   = (const int*)d_in[1];
    // d_in[2] = dep_ids (unused), d_in[5] = adj (unused)
    const int*   subj_pos  = (const int*)d_in[3];
    const int*   obj_pos   = (const int*)d_in[4];
    const float* WQ        = (const float*)d_in[6];
    const float* bQ        = (const float*)d_in[7];
    const float* WK        = (const float*)d_in[8];
    const float* bK        = (const float*)d_in[9];
    const float* WV_w      = (const float*)d_in[10];
    const float* WV_b      = (const float*)d_in[11];
    const float* Wout_w    = (const float*)d_in[12];
    const float* Wout_b    = (const float*)d_in[13];
    const float* ln_g      = (const float*)d_in[14];
    const float* ln_b      = (const float*)d_in[15];
    const float* label_emb = (const float*)d_in[16];
    const float* feat_w    = (const float*)d_in[17];
    const float* feat_b    = (const float*)d_in[18];
    const float* out_w     = (const float*)d_in[19];
    const float* out_b     = (const float*)d_in[20];

    float* out = (float*)d_out;
    float* ws  = (float*)d_ws;

    // workspace layout (floats)
    float* w_eff  = ws;                 // 768
    float* c_val  = ws + 768;           // 1 (pad to 800)
    float* pooled = ws + 800;           // 128*768
    float* feats  = ws + 800 + 98304;                         // 128*2304
    float* tmp1   = ws + 800 + 98304 + 294912;                // 128*768
    float* ctxg   = ws + 800 + 98304 + 294912 + 98304;        // 128*768
    float* h1v    = ws + 800 + 98304 + 294912 + 2 * 98304;    // 128*768
    float* h2v    = ws + 800 + 98304 + 294912 + 3 * 98304;    // 128*300

    prep_kernel<<<1, 256, 0, stream>>>(label_emb, WK, bK, WQ, bQ, w_eff, c_val);

    attn_pool_kernel<<<BB, 256, 0, stream>>>(inputs, pos_ids, subj_pos, obj_pos,
                                             w_eff, c_val, pooled, feats);

    // ctx_pre = pooled @ WV_w^T + WV_b        (128x768x768)
    gemm_wmma_f32<0><<<dim3(12, 8), 128, 0, stream>>>(pooled, WV_w, WV_b, tmp1,
                                                      BB, HH, HH);
    // ctx_g = gelu(ctx_pre @ Wout_w^T + Wout_b)
    gemm_wmma_f32<1><<<dim3(12, 8), 128, 0, stream>>>(tmp1, Wout_w, Wout_b, ctxg,
                                                      BB, HH, HH);
    // layernorm -> d_out ctx section + feats[:,1536:2304]
    ln_kernel<<<BB, 256, 0, stream>>>(ctxg, ln_g, ln_b, out + BB * NLAB, feats);

    // h1 = gelu(feats @ feat_w^T + feat_b)    (128x768, K=2304)
    gemm_wmma_f32<1><<<dim3(12, 8), 128, 0, stream>>>(feats, feat_w, feat_b, h1v,
                                                      BB, HH, FEAT);
    // h2 = gelu(h1 @ out_w^T + out_b)         (128x300, K=768)
    gemm_wmma_f32<1><<<dim3(5, 8), 128, 0, stream>>>(h1v, out_w, out_b, h2v,
                                                     BB, LDIM, HH);
    // logits
    logits_kernel<<<BB, 64, 0, stream>>>(h2v, label_emb, out);
}